// VGAE_39865886441823
// MI455X (gfx1250) — compile-verified
//
#include <hip/hip_runtime.h>
#include <hip/hip_bf16.h>
#include <math.h>

typedef __attribute__((ext_vector_type(2))) float v2f;
typedef __attribute__((ext_vector_type(8))) float v8f;

#define EMB_DIM 64
#define HID_DIM 64
#define OUT_DIM 32

// ---------------------------------------------------------------------------
// deg[i] = 1 (self loop)
__global__ void k_deg_init(float* __restrict__ deg, int n) {
    int i = blockIdx.x * blockDim.x + threadIdx.x;
    if (i < n) deg[i] = 1.0f;
}

// deg[dst] += 1 per edge
__global__ void k_deg_scatter(const int* __restrict__ ei, float* __restrict__ deg, int nedges) {
    int e = blockIdx.x * blockDim.x + threadIdx.x;
    if (e < nedges) atomicAdd(&deg[ei[nedges + e]], 1.0f);
}

// dinv[i] = rsqrt(deg[i])   (deg >= 1 always due to self loop)
__global__ void k_rsqrt(float* __restrict__ d, int n) {
    int i = blockIdx.x * blockDim.x + threadIdx.x;
    if (i < n) d[i] = rsqrtf(d[i]);
}

// ---------------------------------------------------------------------------
// Repack a row-major [K, ncols] weight matrix into WMMA-B fragment order:
// packed[((k0/4)*ntiles + nt)*32 + lane] (v2f) =
//   { W[k0 + 2*(lane>>4) + 0][nt*16 + (lane&15)],
//     W[k0 + 2*(lane>>4) + 1][nt*16 + (lane&15)] }
// One thread per v2f fragment -> B loads in the GEMM become coalesced b64.
__global__ void k_pack_w(const float* __restrict__ W, float* __restrict__ Wp,
                         int K, int ncols) {
    int ntiles = ncols >> 4;
    int total  = (K >> 2) * ntiles * 32;
    int t = blockIdx.x * blockDim.x + threadIdx.x;
    if (t >= total) return;
    int lane = t & 31;
    int nt   = (t >> 5) % ntiles;
    int step = t / (32 * ntiles);
    int col  = nt * 16 + (lane & 15);
    int krow = step * 4 + ((lane >> 4) * 2);
    Wp[2 * t + 0] = W[(size_t)(krow + 0) * ncols + col];
    Wp[2 * t + 1] = W[(size_t)(krow + 1) * ncols + col];
}

// ---------------------------------------------------------------------------
// Embedding bag (weighted sum of 64-dim rows) + L2 row normalize.
// One wave32 per node: lane holds dims (2*lane, 2*lane+1).
__global__ void k_embed(const int* __restrict__ fidx, const int* __restrict__ foff,
                        const float* __restrict__ fw, const float* __restrict__ table,
                        float* __restrict__ x, int nnodes, int total_nnz) {
    int node = blockIdx.x * (blockDim.x >> 5) + (threadIdx.x >> 5);
    int lane = threadIdx.x & 31;
    if (node >= nnodes) return;
    int s = foff[node];
    int e = (node + 1 < nnodes) ? foff[node + 1] : total_nnz;
    float ax = 0.0f, ay = 0.0f;
    for (int j = s; j < e; ++j) {
        int   ix = fidx[j];
        float wt = fw[j];
        const float2 v = *(const float2*)(table + (size_t)ix * EMB_DIM + 2 * lane);
        ax = fmaf(v.x, wt, ax);
        ay = fmaf(v.y, wt, ay);
    }
    // wave-wide sum of squares (wave32)
    float ss = ax * ax + ay * ay;
    #pragma unroll
    for (int m = 16; m >= 1; m >>= 1) ss += __shfl_xor(ss, m, 32);
    float inv = 1.0f / fmaxf(sqrtf(ss), 1e-12f);
    float2 o = {ax * inv, ay * inv};
    *(float2*)(x + (size_t)node * EMB_DIM + 2 * lane) = o;
}

// ---------------------------------------------------------------------------
// agg[i][:] = x[i][:] * dinv[i]^2   (self-loop term, also initializes buffer)
// One float4 per thread.
__global__ void k_agg_init(const float* __restrict__ x, const float* __restrict__ dinv,
                           float* __restrict__ agg, int nnodes) {
    size_t t = (size_t)blockIdx.x * blockDim.x + threadIdx.x;
    if (t >= (size_t)nnodes * (EMB_DIM / 4)) return;
    int node = (int)(t >> 4);
    float s = dinv[node];
    s = s * s;
    float4 v = ((const float4*)x)[t];
    float4 o = {v.x * s, v.y * s, v.z * s, v.w * s};
    ((float4*)agg)[t] = o;
}

// agg[dst][:] += x[src][:] * dinv[src]*dinv[dst]; 16 threads per edge, float4 each.
__global__ void k_agg_edges(const float* __restrict__ x, const int* __restrict__ ei,
                            const float* __restrict__ dinv, float* __restrict__ agg,
                            int nedges) {
    size_t t = (size_t)blockIdx.x * blockDim.x + threadIdx.x;
    int e  = (int)(t >> 4);
    int d4 = ((int)t & 15) * 4;
    if (e >= nedges) return;
    int src = ei[e];
    int dst = ei[nedges + e];
    float norm = dinv[src] * dinv[dst];
    const float4 v = *(const float4*)(x + (size_t)src * EMB_DIM + d4);
    float* o = agg + (size_t)dst * EMB_DIM + d4;
    atomicAdd(o + 0, v.x * norm);
    atomicAdd(o + 1, v.y * norm);
    atomicAdd(o + 2, v.z * norm);
    atomicAdd(o + 3, v.w * norm);
}

// ---------------------------------------------------------------------------
// x1 = relu(agg @ W1 + b1).   agg: [N,64], W1p: pre-packed B fragments.
// One wave per 16-row slab; 4 N-tiles of 16; K stepped by 4 via
// V_WMMA_F32_16X16X4_F32. A-fragment reused across the 4 N-tiles;
// each B fragment is one coalesced global_load_b64 from the packed layout.
__global__ void k_gcn1(const float* __restrict__ agg, const float* __restrict__ W1p,
                       const float* __restrict__ b1, float* __restrict__ x1, int nnodes) {
    int tile = blockIdx.x * (blockDim.x >> 5) + (threadIdx.x >> 5);
    int lane = threadIdx.x & 31;
    int rbase = tile * 16;
    if (rbase >= nnodes) return;

    int mLocal = lane & 15;          // row (A) / col (B,D) within tile
    int kHalf  = (lane >> 4) * 2;    // lanes 16-31 hold K+2,K+3
    int arow   = rbase + mLocal;
    if (arow >= nnodes) arow = nnodes - 1;   // clamp loads on ragged tail

    v8f acc0 = {0.f,0.f,0.f,0.f,0.f,0.f,0.f,0.f};
    v8f acc1 = acc0, acc2 = acc0, acc3 = acc0;

    const float* arowp = agg + (size_t)arow * HID_DIM;
    const v2f*   Bp    = (const v2f*)W1p + lane;   // + (step*4 + nt)*32
    #pragma unroll
    for (int k0 = 0; k0 < HID_DIM; k0 += 4) {
        v2f a = *(const v2f*)(arowp + k0 + kHalf);        // A 16x4 fragment
        int sbase = (k0 >> 2) * 4 * 32;
        v2f b0 = Bp[sbase +  0];
        v2f b1v = Bp[sbase + 32];
        v2f b2 = Bp[sbase + 64];
        v2f b3 = Bp[sbase + 96];
        acc0 = __builtin_amdgcn_wmma_f32_16x16x4_f32(false, a, false, b0, (short)0, acc0, false, false);
        acc1 = __builtin_amdgcn_wmma_f32_16x16x4_f32(false, a, false, b1v, (short)0, acc1, false, false);
        acc2 = __builtin_amdgcn_wmma_f32_16x16x4_f32(false, a, false, b2, (short)0, acc2, false, false);
        acc3 = __builtin_amdgcn_wmma_f32_16x16x4_f32(false, a, false, b3, (short)0, acc3, false, false);
    }

    int rowHalf = (lane >> 4) * 8;   // D: VGPR r -> row r (lanes 0-15) / r+8 (16-31)
    v8f accs[4] = {acc0, acc1, acc2, acc3};
    #pragma unroll
    for (int nt = 0; nt < 4; ++nt) {
        int col = nt * 16 + mLocal;
        float bias = b1[col];
        #pragma unroll
        for (int r = 0; r < 8; ++r) {
            int row = rbase + rowHalf + r;
            if (row < nnodes)
                x1[(size_t)row * HID_DIM + col] = fmaxf(accs[nt][r] + bias, 0.0f);
        }
    }
}

// ---------------------------------------------------------------------------
// z = (agg2 @ Wmu + bmu) + noise * exp(agg2 @ Wls + bls)
// Wmup/Wlsp: pre-packed B fragments (2 N-tiles each). A reused 4x.
__global__ void k_out(const float* __restrict__ agg, const float* __restrict__ Wmup,
                      const float* __restrict__ bmu, const float* __restrict__ Wlsp,
                      const float* __restrict__ bls, const float* __restrict__ noise,
                      float* __restrict__ z, int nnodes) {
    int tile = blockIdx.x * (blockDim.x >> 5) + (threadIdx.x >> 5);
    int lane = threadIdx.x & 31;
    int rbase = tile * 16;
    if (rbase >= nnodes) return;

    int mLocal = lane & 15;
    int kHalf  = (lane >> 4) * 2;
    int arow   = rbase + mLocal;
    if (arow >= nnodes) arow = nnodes - 1;

    v8f mu0 = {0.f,0.f,0.f,0.f,0.f,0.f,0.f,0.f};
    v8f mu1 = mu0, ls0 = mu0, ls1 = mu0;

    const float* arowp = agg + (size_t)arow * HID_DIM;
    const v2f*   Bmu   = (const v2f*)Wmup + lane;   // + (step*2 + nt)*32
    const v2f*   Bls   = (const v2f*)Wlsp + lane;
    #pragma unroll
    for (int k0 = 0; k0 < HID_DIM; k0 += 4) {
        v2f a = *(const v2f*)(arowp + k0 + kHalf);
        int sbase = (k0 >> 2) * 2 * 32;
        v2f bm0 = Bmu[sbase +  0];
        v2f bm1 = Bmu[sbase + 32];
        v2f bl0 = Bls[sbase +  0];
        v2f bl1 = Bls[sbase + 32];
        mu0 = __builtin_amdgcn_wmma_f32_16x16x4_f32(false, a, false, bm0, (short)0, mu0, false, false);
        mu1 = __builtin_amdgcn_wmma_f32_16x16x4_f32(false, a, false, bm1, (short)0, mu1, false, false);
        ls0 = __builtin_amdgcn_wmma_f32_16x16x4_f32(false, a, false, bl0, (short)0, ls0, false, false);
        ls1 = __builtin_amdgcn_wmma_f32_16x16x4_f32(false, a, false, bl1, (short)0, ls1, false, false);
    }

    int rowHalf = (lane >> 4) * 8;
    v8f mus[2] = {mu0, mu1};
    v8f lss[2] = {ls0, ls1};
    #pragma unroll
    for (int nt = 0; nt < 2; ++nt) {
        int col = nt * 16 + mLocal;
        float bm = bmu[col];
        float bl = bls[col];
        #pragma unroll
        for (int r = 0; r < 8; ++r) {
            int row = rbase + rowHalf + r;
            if (row < nnodes) {
                size_t o = (size_t)row * OUT_DIM + col;
                float mu = mus[nt][r] + bm;
                float ls = lss[nt][r] + bl;
                z[o] = fmaf(noise[o], expf(ls), mu);
            }
        }
    }
}

// ---------------------------------------------------------------------------
extern "C" void kernel_launch(void* const* d_in, const int* in_sizes, int n_in,
                              void* d_out, int out_size, void* d_ws, size_t ws_size,
                              hipStream_t stream) {
    const int*   fidx  = (const int*)d_in[0];
    const int*   foff  = (const int*)d_in[1];
    const float* fw    = (const float*)d_in[2];
    const int*   eidx  = (const int*)d_in[3];
    const float* noise = (const float*)d_in[4];
    const float* table = (const float*)d_in[5];
    const float* W1    = (const float*)d_in[6];
    const float* b1    = (const float*)d_in[7];
    const float* Wmu   = (const float*)d_in[8];
    const float* bmu   = (const float*)d_in[9];
    const float* Wls   = (const float*)d_in[10];
    const float* bls   = (const float*)d_in[11];

    const int N         = in_sizes[1];       // N_NODES
    const int total_nnz = in_sizes[0];
    const int E         = in_sizes[3] / 2;   // N_EDGES

    // workspace: x | agg | x1 | dinv | packed weights ; agg2 reuses x's slot
    float* xbuf   = (float*)d_ws;
    float* aggbuf = xbuf   + (size_t)N * EMB_DIM;
    float* x1buf  = aggbuf + (size_t)N * EMB_DIM;
    float* dinv   = x1buf  + (size_t)N * EMB_DIM;
    float* W1p    = dinv   + (size_t)N;            // 64*64 floats
    float* Wmup   = W1p    + 64 * 64;              // 64*32 floats
    float* Wlsp   = Wmup   + 64 * 32;              // 64*32 floats
    float* agg2   = xbuf;  // reuse

    const int T = 256;

    // one-time weight repack into WMMA B-fragment order
    k_pack_w<<<(2048 + T - 1) / T, T, 0, stream>>>(W1,  W1p,  HID_DIM, HID_DIM);
    k_pack_w<<<(1024 + T - 1) / T, T, 0, stream>>>(Wmu, Wmup, HID_DIM, OUT_DIM);
    k_pack_w<<<(1024 + T - 1) / T, T, 0, stream>>>(Wls, Wlsp, HID_DIM, OUT_DIM);

    // degrees -> dinv
    k_deg_init<<<(N + T - 1) / T, T, 0, stream>>>(dinv, N);
    k_deg_scatter<<<(E + T - 1) / T, T, 0, stream>>>(eidx, dinv, E);
    k_rsqrt<<<(N + T - 1) / T, T, 0, stream>>>(dinv, N);

    // x = normalize(embedding_bag)
    k_embed<<<(N + 7) / 8, T, 0, stream>>>(fidx, foff, fw, table, xbuf, N, total_nnz);

    // agg1 = D^-1/2 (A+I) D^-1/2 x
    {
        size_t work = (size_t)N * (EMB_DIM / 4);
        k_agg_init<<<(unsigned)((work + T - 1) / T), T, 0, stream>>>(xbuf, dinv, aggbuf, N);
        size_t ework = (size_t)E * 16;
        k_agg_edges<<<(unsigned)((ework + T - 1) / T), T, 0, stream>>>(xbuf, eidx, dinv, aggbuf, E);
    }

    // x1 = relu(agg1 @ W1 + b1)     [WMMA f32 16x16x4]
    k_gcn1<<<(N / 16 + 7) / 8, T, 0, stream>>>(aggbuf, W1p, b1, x1buf, N);

    // agg2 = D^-1/2 (A+I) D^-1/2 x1   (shared between mu and logstd convs)
    {
        size_t work = (size_t)N * (EMB_DIM / 4);
        k_agg_init<<<(unsigned)((work + T - 1) / T), T, 0, stream>>>(x1buf, dinv, agg2, N);
        size_t ework = (size_t)E * 16;
        k_agg_edges<<<(unsigned)((ework + T - 1) / T), T, 0, stream>>>(x1buf, eidx, dinv, agg2, E);
    }

    // z = (agg2@Wmu+bmu) + noise*exp(agg2@Wls+bls)   [WMMA f32 16x16x4]
    k_out<<<(N / 16 + 7) / 8, T, 0, stream>>>(agg2, Wmup, bmu, Wlsp, bls, noise, (float*)d_out, N);
}